// Render_56702158242193
// MI455X (gfx1250) — compile-verified
//
#include <hip/hip_runtime.h>
#include <hip/hip_bf16.h>
#include <math.h>

typedef __attribute__((ext_vector_type(2))) float v2f;
typedef __attribute__((ext_vector_type(8))) float v8f;

// ws layout (in floats):
#define WSA_OFF 0        // 64 groups * 64 floats  = 4096 floats (WMMA A-tiles)
#define WSS_OFF 4096     // 256 tris  * 16 floats  = 4096 floats (shading records)
#define WSZ_OFF 8192     // 1 float (zmin)
#define NGROUP  64       // 256 triangles / 4 per WMMA group

// ---------------- Kernel 1: per-triangle setup (1 block, 256 threads) -------
__global__ __launch_bounds__(256)
void setup_kernel(const float* __restrict__ vertices, const int* __restrict__ faces,
                  const float* __restrict__ uv, const int* __restrict__ uvfaces,
                  float* __restrict__ ws) {
  __shared__ float red[256];
  const int t = threadIdx.x;

  // zinit = min z over 1024 vertices
  float zm = __builtin_inff();
  #pragma unroll
  for (int k = 0; k < 4; ++k)
    zm = fminf(zm, vertices[3*(t*4 + k) + 2]);
  red[t] = zm;
  __syncthreads();
  for (int s = 128; s > 0; s >>= 1) {
    if (t < s) red[t] = fminf(red[t], red[t + s]);
    __syncthreads();
  }
  if (t == 0) ws[WSZ_OFF] = red[0];

  // triangle t
  const int i0 = faces[3*t+0], i1 = faces[3*t+1], i2 = faces[3*t+2];
  const float ax = vertices[3*i0+0], ay = vertices[3*i0+1], az = vertices[3*i0+2];
  const float bx = vertices[3*i1+0], by = vertices[3*i1+1], bz = vertices[3*i1+2];
  const float cx = vertices[3*i2+0], cy = vertices[3*i2+1], cz = vertices[3*i2+2];

  // 2D signed area == face-normal z (scalar normalization cancels; only sign matters)
  const float A = (bx-ax)*(cy-ay) - (by-ay)*(cx-ax);
  const bool  valid = (A > 0.0f) && (A >= 1e-9f);
  const float Asafe = (fabsf(A) < 1e-9f) ? 1.0f : A;
  const float z0i = 1.0f/az, z1i = 1.0f/bz, z2i = 1.0f/cz;

  // edge(a,c) = (px-ax)(cy-ay) - (py-ay)(cx-ax) -> affine in (px,py)
  // pAB: a=v1,c=v0 ; pCB: a=v2,c=v1 ; pCA: a=v0,c=v2
  float e0x = ay - by, e0y = -(ax - bx), e0c = -bx*(ay-by) + by*(ax-bx);
  float e1x = by - cy, e1y = -(bx - cx), e1c = -cx*(by-cy) + cy*(bx-cx);
  float e2x = cy - ay, e2y = -(cx - ax), e2c = -ax*(cy-ay) + ay*(cx-ax);
  // zinv(p) = z2i + (pCB*(z0i-z2i) + pCA*(z1i-z2i)) / Asafe
  const float inva = 1.0f / Asafe;
  const float zx = (e1x*(z0i - z2i) + e2x*(z1i - z2i)) * inva;
  const float zy = (e1y*(z0i - z2i) + e2y*(z1i - z2i)) * inva;
  const float zc = (e1c*(z0i - z2i) + e2c*(z1i - z2i)) * inva + z2i;

  // shading record (16 floats) — written with the REAL edges
  {
    const int u0 = uvfaces[3*t+0], u1 = uvfaces[3*t+1], u2 = uvfaces[3*t+2];
    float* r = ws + WSS_OFF + t*16;
    r[0] = Asafe;
    r[1] = e1x; r[2] = e1y; r[3] = e1c;          // pCB -> w1
    r[4] = e2x; r[5] = e2y; r[6] = e2c;          // pCA -> w2
    r[7]  = (uv[2*u0+0]*2.0f - 1.0f)*z0i; r[8]  = (uv[2*u0+1]*2.0f - 1.0f)*z0i;
    r[9]  = (uv[2*u1+0]*2.0f - 1.0f)*z1i; r[10] = (uv[2*u1+1]*2.0f - 1.0f)*z1i;
    r[11] = (uv[2*u2+0]*2.0f - 1.0f)*z2i; r[12] = (uv[2*u2+1]*2.0f - 1.0f)*z2i;
    r[13] = z0i; r[14] = z1i; r[15] = z2i;
  }

  // fold validity into the edge functions: invalid -> constant -1 (never covers)
  if (!valid) {
    e0x = 0.0f; e0y = 0.0f; e0c = -1.0f;
    e1x = 0.0f; e1y = 0.0f; e1c = -1.0f;
    e2x = 0.0f; e2y = 0.0f; e2c = -1.0f;
  }

  // WMMA A-tile coefficients. Group g = t/4; func row L = 4*(t%4)+f.
  // lanes 0..15 of A hold (K0,K1) = (coef_x, coef_y); lanes 16..31 hold (K2,K3) = (const, 0)
  const float fx[4] = {e0x, e1x, e2x, zx};
  const float fy[4] = {e0y, e1y, e2y, zy};
  const float fc[4] = {e0c, e1c, e2c, zc};
  float* base = ws + WSA_OFF + (t >> 2)*64;
  const int l0 = (t & 3) * 4;
  #pragma unroll
  for (int f = 0; f < 4; ++f) {
    const int L = l0 + f;
    base[L*2 + 0]      = fx[f];
    base[L*2 + 1]      = fy[f];
    base[32 + L*2 + 0] = fc[f];
    base[32 + L*2 + 1] = 0.0f;
  }
}

// ---------------- Kernel 2: WMMA raster + z-argmin + shade ------------------
// Each wave rasterizes 32 pixels (two 16-pixel tiles) against all 256 triangles.
// min(zinv) tracking (no divides); z-clip folded into the argmin init value.
__global__ __launch_bounds__(256)
void raster_kernel(const float* __restrict__ ws, const float* __restrict__ uvmap,
                   float* __restrict__ out) {
  __shared__ float scoef[4096];           // 16 KB coefficient table
  {
    const float4* src = (const float4*)(ws + WSA_OFF);
    float4* dst = (float4*)scoef;
    #pragma unroll
    for (int k = 0; k < 4; ++k)
      dst[threadIdx.x + k*256] = src[threadIdx.x + k*256];
  }
  __builtin_prefetch(ws + WSS_OFF + threadIdx.x*16, 0, 0);  // warm shading records
  __syncthreads();

  const int gtid = blockIdx.x * 256 + threadIdx.x;
  const int wave = gtid >> 5;            // 0..2047, 32 pixels per wave
  const int lane = threadIdx.x & 31;
  const int hf   = lane >> 4;
  const int l16  = lane & 15;

  const int pbase = wave * 32;           // flat pixel base (same image row)
  const int row   = pbase >> 8;
  const int jbase = pbase & 255;

  const float step = 2.0f / 255.0f;
  const float py  = 1.0f - (float)row * step;
  const float px0 = -1.0f + (float)(jbase + l16) * step;
  const float px1 = px0 + 16.0f * step;

  // B 4x16: rows (px, py, 1, 0); lanes 0-15 hold K0/K1, lanes 16-31 hold K2/K3
  v2f b0, b1;
  b0.x = hf ? 1.0f : px0;  b0.y = hf ? 0.0f : py;
  b1.x = hf ? 1.0f : px1;  b1.y = hf ? 0.0f : py;

  // zi <= zlim && zi < bz  ===  zi < bzInit  with  bzInit = nextafter(zlim,+inf)
  const float zlim = 1.0f / ws[WSZ_OFF];        // 1/zinit, positive normal
  const float bzInit = __uint_as_float(__float_as_uint(zlim) + 1u);

  float bz0 = bzInit, bz1 = bzInit;
  int   bi0 = 0,      bi1 = 0;

  // software double-buffer the A-tile so ds_load latency overlaps WMMA+compares
  v2f a = *(const v2f*)(scoef + lane*2);
  for (int g = 0; g < NGROUP; ++g) {
    const int gn = (g + 1) & (NGROUP - 1);
    const v2f an = *(const v2f*)(scoef + gn*64 + lane*2);

    v8f cz = {};
    const v8f d0 = __builtin_amdgcn_wmma_f32_16x16x4_f32(
        false, a, false, b0, (short)0, cz, false, false);
    const v8f d1 = __builtin_amdgcn_wmma_f32_16x16x4_f32(
        false, a, false, b1, (short)0, cz, false, false);

    const int triBase = g*4 + hf*2;     // this lane's 2 triangles
    #pragma unroll
    for (int j = 0; j < 2; ++j) {
      {
        const float m  = __builtin_fminf(__builtin_fminf(d0[4*j+0], d0[4*j+1]), d0[4*j+2]);
        const float zi = d0[4*j+3];
        const bool upd = (m > 0.0f) & (zi < bz0);
        bz0 = upd ? zi : bz0;
        bi0 = upd ? (triBase + j) : bi0;
      }
      {
        const float m  = __builtin_fminf(__builtin_fminf(d1[4*j+0], d1[4*j+1]), d1[4*j+2]);
        const float zi = d1[4*j+3];
        const bool upd = (m > 0.0f) & (zi < bz1);
        bz1 = upd ? zi : bz1;
        bi1 = upd ? (triBase + j) : bi1;
      }
    }
    a = an;
  }

  // merge the two wave-halves (same pixel, different triangle subsets);
  // tie-break on lower index to match argmax first-max semantics
  {
    const float oZ = __shfl_xor(bz0, 16, 32);
    const int   oI = __shfl_xor(bi0, 16, 32);
    const bool take = (oZ < bz0) || ((oZ == bz0) && (oI < bi0));
    bz0 = take ? oZ : bz0;  bi0 = take ? oI : bi0;
  }
  {
    const float oZ = __shfl_xor(bz1, 16, 32);
    const int   oI = __shfl_xor(bi1, 16, 32);
    const bool take = (oZ < bz1) || ((oZ == bz1) && (oI < bi1));
    bz1 = take ? oZ : bz1;  bi1 = take ? oI : bi1;
  }

  // lanes 0-15 shade tile 0's pixel, lanes 16-31 shade tile 1's pixel
  const float bz  = hf ? bz1 : bz0;
  const int   bi  = hf ? bi1 : bi0;
  const float pxs = hf ? px1 : px0;
  const bool  hit = bz < bzInit;

  const float* r = ws + WSS_OFF + bi * 16;
  const float ia = 1.0f / r[0];
  const float w1 = (r[1]*pxs + r[2]*py + r[3]) * ia;
  const float w2 = (r[4]*pxs + r[5]*py + r[6]) * ia;
  const float w3 = 1.0f - w1 - w2;
  const float su  = w1*r[7]  + w2*r[9]  + w3*r[11];
  const float sv  = w1*r[8]  + w2*r[10] + w3*r[12];
  const float szi = w1*r[13] + w2*r[14] + w3*r[15];
  const float Zw  = 1.0f / (hit ? szi : 1.0f);
  const float gx = su * Zw, gy = sv * Zw;

  // bilinear sample of 3x256x256 with border-zero
  const float x = ((gx + 1.0f)*256.0f - 1.0f)*0.5f;
  const float y = ((gy + 1.0f)*256.0f - 1.0f)*0.5f;
  const float x0 = floorf(x), y0 = floorf(y);
  const float wx1 = x - x0, wx0 = 1.0f - wx1;
  const float wy1 = y - y0, wy0 = 1.0f - wy1;

  float rgb[3] = {0.0f, 0.0f, 0.0f};
  #pragma unroll
  for (int ty = 0; ty < 2; ++ty) {
    const float yy = y0 + (float)ty;
    const bool vy = (yy >= 0.0f) && (yy <= 255.0f);
    const int  iy = (int)fminf(fmaxf(yy, 0.0f), 255.0f);
    const float wy = ty ? wy1 : wy0;
    #pragma unroll
    for (int tx = 0; tx < 2; ++tx) {
      const float xx = x0 + (float)tx;
      const bool vx = (xx >= 0.0f) && (xx <= 255.0f);
      const int  ix = (int)fminf(fmaxf(xx, 0.0f), 255.0f);
      const float wgt = ((vx && vy) ? 1.0f : 0.0f) * wy * (tx ? wx1 : wx0);
      #pragma unroll
      for (int c = 0; c < 3; ++c)
        rgb[c] += uvmap[c*65536 + iy*256 + ix] * wgt;
    }
  }

  const float hitf = hit ? 1.0f : 0.0f;
  const int p = pbase + hf*16 + l16;
  out[0*65536 + p] = rgb[0]*hitf;
  out[1*65536 + p] = rgb[1]*hitf;
  out[2*65536 + p] = rgb[2]*hitf;
  out[3*65536 + p] = hitf;
}

extern "C" void kernel_launch(void* const* d_in, const int* in_sizes, int n_in,
                              void* d_out, int out_size, void* d_ws, size_t ws_size,
                              hipStream_t stream) {
  const float* vertices = (const float*)d_in[0];   // (1024,3) f32
  const int*   faces    = (const int*)d_in[1];     // (256,3)
  const float* uv       = (const float*)d_in[2];   // (1024,2) f32
  const int*   uvfaces  = (const int*)d_in[3];     // (256,3)
  const float* uvmap    = (const float*)d_in[4];   // (3,256,256) f32
  float* ws  = (float*)d_ws;
  float* out = (float*)d_out;

  setup_kernel<<<1, 256, 0, stream>>>(vertices, faces, uv, uvfaces, ws);
  raster_kernel<<<256, 256, 0, stream>>>(ws, uvmap, out);
}